// DotProductPredictor_8710193677020
// MI455X (gfx1250) — compile-verified
//
#include <hip/hip_runtime.h>
#include <hip/hip_bf16.h>

// CDNA5 / gfx1250: fp32 WMMA 16x16x4 used as a batched-dot engine.
// 16 edges per wave32: C += A(16x4) * B(4x16), diag(C) = per-edge dots.

typedef float v2f __attribute__((ext_vector_type(2)));
typedef float v8f __attribute__((ext_vector_type(8)));

#define D_FEAT 128

__global__ __launch_bounds__(256) void edge_dot_wmma_f32(
    const float* __restrict__ x,
    const int*   __restrict__ senders,
    const int*   __restrict__ receivers,
    float*       __restrict__ he,
    int n_edges)
{
    const int lane  = threadIdx.x & 31;
    const int wave  = threadIdx.x >> 5;
    const int group = blockIdx.x * (blockDim.x >> 5) + wave;   // 16 edges per wave
    const int ebase = group << 4;
    if (ebase >= n_edges) return;   // wave-uniform exit: EXEC stays all-ones below

    const int lp   = lane & 15;    // edge-in-group this lane serves
    const int half = lane >> 4;    // K-halves per ISA 16x4 / 4x16 fp32 striping

    // Clamp tail edges so every lane issues in-bounds loads (EXEC must be ~0
    // for WMMA); invalid lanes simply don't store.
    int e  = ebase + lp;
    int ec = (e < n_edges) ? e : (n_edges - 1);

    const long long srow = (long long)senders[ec];
    const long long rrow = (long long)receivers[ec];

    // A (16x4): lane<16 holds K = {k, k+1}; lane>=16 holds K = {k+2, k+3}
    // B (4x16): identical per-lane float2 pattern from the receiver row.
    const float* __restrict__ ap = x + srow * D_FEAT + (half << 1);
    const float* __restrict__ bp = x + rrow * D_FEAT + (half << 1);

    v8f c0 = {};
    v8f c1 = {};

#pragma unroll
    for (int k = 0; k < D_FEAT; k += 8) {
        v2f a0 = *(const v2f*)(ap + k);
        v2f b0 = *(const v2f*)(bp + k);
        v2f a1 = *(const v2f*)(ap + k + 4);
        v2f b1 = *(const v2f*)(bp + k + 4);
        // (neg_a, A, neg_b, B, c_mod, C, reuse_a, reuse_b)
        c0 = __builtin_amdgcn_wmma_f32_16x16x4_f32(false, a0, false, b0,
                                                   (short)0, c0, false, false);
        c1 = __builtin_amdgcn_wmma_f32_16x16x4_f32(false, a1, false, b1,
                                                   (short)0, c1, false, false);
    }

    // Diagonal extraction (C layout: lanes 0-15 rows M=v, lanes 16-31 rows M=v+8,
    // column N = lane&15). Edge m<8 -> lane m, VGPR m; edge m>=8 -> lane m+16,
    // VGPR m-8.
    int want = -1;
    if (half == 0 && lp < 8)  want = lp;
    if (half == 1 && lp >= 8) want = lp - 8;

    float d = 0.0f;
#pragma unroll
    for (int i = 0; i < 8; ++i) {
        if (want == i) d = c0[i] + c1[i];
    }

    if (want >= 0) {
        int eo = ebase + lp;
        if (eo < n_edges) he[eo] = d;
    }
}

extern "C" void kernel_launch(void* const* d_in, const int* in_sizes, int n_in,
                              void* d_out, int out_size, void* d_ws, size_t ws_size,
                              hipStream_t stream)
{
    const float* x  = (const float*)d_in[0];
    const int*   ei = (const int*)d_in[1];      // harness contract: integer -> int32
    const int E = in_sizes[1] / 2;              // edge_index is [2, E] flattened
    const int* senders   = ei;
    const int* receivers = ei + E;
    float* he = (float*)d_out;

    const int waves_per_block = 8;              // 256 threads = 8 wave32
    const int groups = (E + 15) / 16;
    const int blocks = (groups + waves_per_block - 1) / waves_per_block;

    hipLaunchKernelGGL(edge_dot_wmma_f32, dim3(blocks), dim3(256), 0, stream,
                       x, senders, receivers, he, E);
}